// Gating_45354854646406
// MI455X (gfx1250) — compile-verified
//
#include <hip/hip_runtime.h>
#include <hip/hip_bf16.h>
#include <math.h>

typedef __attribute__((ext_vector_type(2))) float v2f;
typedef __attribute__((ext_vector_type(4))) float v4f;
typedef __attribute__((ext_vector_type(8))) float v8f;
typedef __attribute__((ext_vector_type(4))) int   v4i;

#define TOKENS 16384   // B*S = 4*4096
#define HDIM   4096
#define NEXP   64
#define KTOP   8
#define BM     128     // tokens per workgroup
#define KT     64      // K-chunk
#define NCHUNK (HDIM / KT)
#define LDP    (KT + 4) // padded LDS row (68 floats; 272 B = 17*16 keeps 16B align)

// ---- CDNA5 async global->LDS path (ASYNCcnt), with portable fallback ------
#if defined(__has_builtin)
#  if __has_builtin(__builtin_amdgcn_global_load_async_to_lds_b128)
#    define HAVE_ASYNC_LDS 1
#  endif
#endif
#ifndef HAVE_ASYNC_LDS
#  define HAVE_ASYNC_LDS 0
#endif

typedef __attribute__((address_space(1))) v4i gv4i; // global int4
typedef __attribute__((address_space(3))) v4i lv4i; // LDS int4

__device__ __forceinline__ void copy16_to_lds(const float* g, float* l) {
#if HAVE_ASYNC_LDS
    // copies 16 B per lane from the global address g to the LDS address l, no VGPR round trip; tracked by ASYNCcnt
    __builtin_amdgcn_global_load_async_to_lds_b128((gv4i*)g, (lv4i*)l,
                                                   /*offset=*/0, /*cpol=*/0);
#else
    *(v4f*)l = *(const v4f*)g;
#endif
}

__device__ __forceinline__ void wait_async_copies() {
#if HAVE_ASYNC_LDS
#  if __has_builtin(__builtin_amdgcn_s_wait_asynccnt)
    __builtin_amdgcn_s_wait_asynccnt(0);
#  else
    asm volatile("s_wait_asynccnt 0x0" ::: "memory");
#  endif
#endif
}

// ---------------------------------------------------------------------------
// Kernel 1: logits[t][e] = sum_k x[t][k] * W[e][k] + b[e]
// 8 waves/block; each wave computes a 16x64 tile via V_WMMA_F32_16X16X4_F32.
// Double-buffered LDS; next chunk streamed via async global->LDS while the
// current chunk's 64 WMMAs execute.
// ---------------------------------------------------------------------------
__global__ __launch_bounds__(256) void router_gemm(
    const float* __restrict__ x,
    const float* __restrict__ W,
    const float* __restrict__ bias,
    float* __restrict__ logits)
{
    __shared__ float xs[2][BM][LDP];   // 2 x 34 KB
    __shared__ float ws[2][NEXP][LDP]; // 2 x 17 KB

    const int tid   = threadIdx.x;
    const int wave  = tid >> 5;
    const int lane  = tid & 31;
    const int hl    = lane >> 4;    // half-wave: 0 or 1
    const int l16   = lane & 15;
    const int mbase = blockIdx.x * BM;

    const int colb = (tid & 15) * 4; // float4 column within chunk
    const int rowb = tid >> 4;       // 0..15

    v8f acc0 = {}, acc1 = {}, acc2 = {}, acc3 = {};

    // issue async copies for one K-chunk into LDS buffer `buf`
    auto stage = [&](int buf, int k0) {
        #pragma unroll
        for (int p = 0; p < 8; ++p) {
            const int row = rowb + p * 16;
            copy16_to_lds(x + (size_t)(mbase + row) * HDIM + k0 + colb,
                          &xs[buf][row][colb]);
        }
        #pragma unroll
        for (int p = 0; p < 4; ++p) {
            const int row = rowb + p * 16;
            copy16_to_lds(W + (size_t)row * HDIM + k0 + colb,
                          &ws[buf][row][colb]);
        }
    };

    // 16 K-steps x 4 N-tiles of V_WMMA_F32_16X16X4_F32 from LDS buffer `buf`
    auto compute = [&](int buf) {
        const int arow = wave * 16 + l16;
        #pragma unroll
        for (int kk = 0; kk < KT / 4; ++kk) {
            const int kc = kk * 4 + 2 * hl;
            // A 16x4 f32: lane = M row; VGPR0/1 = {K0,K1}(lanes0-15)/{K2,K3}(lanes16-31)
            v2f a  = *(const v2f*)&xs[buf][arow][kc];
            // B 4x16 f32: lane = N col; same K striping across the two VGPRs
            v2f b0 = *(const v2f*)&ws[buf][l16 +  0][kc];
            v2f b1 = *(const v2f*)&ws[buf][l16 + 16][kc];
            v2f b2 = *(const v2f*)&ws[buf][l16 + 32][kc];
            v2f b3 = *(const v2f*)&ws[buf][l16 + 48][kc];
            acc0 = __builtin_amdgcn_wmma_f32_16x16x4_f32(false, a, false, b0, (short)0, acc0, false, false);
            acc1 = __builtin_amdgcn_wmma_f32_16x16x4_f32(false, a, false, b1, (short)0, acc1, false, false);
            acc2 = __builtin_amdgcn_wmma_f32_16x16x4_f32(false, a, false, b2, (short)0, acc2, false, false);
            acc3 = __builtin_amdgcn_wmma_f32_16x16x4_f32(false, a, false, b3, (short)0, acc3, false, false);
        }
    };

    // ---- pipelined main loop ----
    stage(0, 0);
    wait_async_copies();
    __syncthreads();

    for (int it = 0; it < NCHUNK; ++it) {
        const int cur = it & 1;
        const int nxt = cur ^ 1;
        if (it + 1 < NCHUNK) stage(nxt, (it + 1) * KT); // overlap with compute
        compute(cur);
        wait_async_copies();   // my async writes into `nxt` landed in LDS
        __syncthreads();       // everyone done reading `cur`, writing `nxt`
    }

    // ---- epilogue: C/D layout (VGPR i: lanes0-15 -> M=i, lanes16-31 -> M=8+i) ----
    const int token0 = mbase + wave * 16 + 8 * hl;
    const float bv0 = bias[l16 +  0];
    const float bv1 = bias[l16 + 16];
    const float bv2 = bias[l16 + 32];
    const float bv3 = bias[l16 + 48];
    #pragma unroll
    for (int i = 0; i < 8; ++i) {
        const size_t base = (size_t)(token0 + i) * NEXP + l16;
        logits[base +  0] = acc0[i] + bv0;
        logits[base + 16] = acc1[i] + bv1;
        logits[base + 32] = acc2[i] + bv2;
        logits[base + 48] = acc3[i] + bv3;
    }
}

// ---------------------------------------------------------------------------
// Kernel 2: per-token top-8 + masked softmax. One wave32 per token,
// 2 experts per lane. Tie-break matches jax.lax.top_k (smaller index first).
// ---------------------------------------------------------------------------
__global__ __launch_bounds__(256) void topk_softmax(
    const float* __restrict__ logits,
    float* __restrict__ sparse,
    int* __restrict__ indices)
{
    const int lane  = threadIdx.x & 31;
    const int token = blockIdx.x * 8 + (threadIdx.x >> 5);

    const float v0o = logits[(size_t)token * NEXP + lane];
    const float v1o = logits[(size_t)token * NEXP + lane + 32];
    float v0 = v0o, v1 = v1o;
    bool s0 = false, s1 = false;

    float topv[KTOP];
    int   topi[KTOP];

    #pragma unroll
    for (int k = 0; k < KTOP; ++k) {
        float mv; int mi;
        if (v0 >= v1) { mv = v0; mi = lane; }       // lane < lane+32: v0 wins ties
        else          { mv = v1; mi = lane + 32; }
        #pragma unroll
        for (int off = 16; off > 0; off >>= 1) {
            const float ov = __shfl_xor(mv, off, 32);
            const int   oi = __shfl_xor(mi, off, 32);
            if (ov > mv || (ov == mv && oi < mi)) { mv = ov; mi = oi; }
        }
        topv[k] = mv; topi[k] = mi;
        if (mi == lane)      { v0 = -INFINITY; s0 = true; }
        if (mi == lane + 32) { v1 = -INFINITY; s1 = true; }
    }

    const float mx = topv[0];
    float denom = 0.0f;
    #pragma unroll
    for (int k = 0; k < KTOP; ++k) denom += __expf(topv[k] - mx);
    const float inv = 1.0f / denom;

    sparse[(size_t)token * NEXP + lane]      = s0 ? __expf(v0o - mx) * inv : 0.0f;
    sparse[(size_t)token * NEXP + lane + 32] = s1 ? __expf(v1o - mx) * inv : 0.0f;

    // lane k grabs topi[k] without a dynamic register index
    int myidx = 0;
    #pragma unroll
    for (int k = 0; k < KTOP; ++k) if (lane == k) myidx = topi[k];
    if (lane < KTOP) indices[(size_t)token * KTOP + lane] = myidx;
}

// ---------------------------------------------------------------------------
extern "C" void kernel_launch(void* const* d_in, const int* in_sizes, int n_in,
                              void* d_out, int out_size, void* d_ws, size_t ws_size,
                              hipStream_t stream)
{
    const float* x  = (const float*)d_in[0];
    const float* W  = (const float*)d_in[1];
    const float* b  = (const float*)d_in[2];

    float* out     = (float*)d_out;
    float* sparse  = out;                                                 // [B,S,E]
    int*   indices = (int*)(out + (size_t)TOKENS * NEXP);                 // [B,S,8] i32
    float* gate    = out + (size_t)TOKENS * NEXP + (size_t)TOKENS * KTOP; // [B*S,E]

    router_gemm<<<TOKENS / BM, 256, 0, stream>>>(x, W, b, gate);
    topk_softmax<<<TOKENS / 8, 256, 0, stream>>>(gate, sparse, indices);
}